// SparseDownsample_69965017252010
// MI455X (gfx1250) — compile-verified
//
#include <hip/hip_runtime.h>
#include <stdint.h>

// ---------------- problem constants (match reference) ----------------
#define N_VOX   524288          // 1<<19
#define CIN     64
#define COUT    128
#define T_DIM   8
#define H_DIM   512
#define W_DIM   512
#define EPS_LN  1e-5f

// ---------------- radix sort config ----------------
#define NBLK    256             // sort blocks per pass
#define TILE    2048            // elements per sort block (N_VOX / NBLK)
#define IPT     8               // items per thread (256 threads)

typedef float v2f __attribute__((ext_vector_type(2)));
typedef float v8f __attribute__((ext_vector_type(8)));

// =====================================================================
// Step 0: linear keys + identity permutation
// =====================================================================
__global__ void lin_init_kernel(const int* __restrict__ indices,
                                uint32_t* __restrict__ lin,
                                uint32_t* __restrict__ valA) {
    int i = blockIdx.x * blockDim.x + threadIdx.x;
    const int* p = indices + 4 * (size_t)i;
    uint32_t b = (uint32_t)p[0], t = (uint32_t)p[1];
    uint32_t y = (uint32_t)p[2], x = (uint32_t)p[3];
    lin[i] = ((b * T_DIM + t) * H_DIM + y) * W_DIM + x;   // 23-bit key
    valA[i] = (uint32_t)i;
}

// =====================================================================
// Radix pass: per-block digit histograms, digit-major layout
// =====================================================================
__global__ void radix_hist_kernel(const uint32_t* __restrict__ lin,
                                  const uint32_t* __restrict__ vals,
                                  uint32_t* __restrict__ gHist, int shift) {
    __shared__ uint32_t h[256];
    h[threadIdx.x] = 0;
    __syncthreads();
    int base = blockIdx.x * TILE;
    for (int i = threadIdx.x; i < TILE; i += 256) {
        uint32_t v = vals[base + i];
        uint32_t d = (lin[v] >> shift) & 255u;
        atomicAdd(&h[d], 1u);
    }
    __syncthreads();
    gHist[threadIdx.x * NBLK + blockIdx.x] = h[threadIdx.x];
}

// =====================================================================
// Global exclusive scan over 256*NBLK = 65536 counters (one block)
// =====================================================================
__global__ void radix_scan_kernel(uint32_t* __restrict__ gHist) {
    __shared__ uint32_t sums[256];
    int t = threadIdx.x;
    int base = t * 256;
    uint32_t s = 0;
    for (int i = 0; i < 256; ++i) s += gHist[base + i];
    sums[t] = s;
    __syncthreads();
    if (t == 0) {
        uint32_t acc = 0;
        for (int i = 0; i < 256; ++i) { uint32_t x = sums[i]; sums[i] = acc; acc += x; }
    }
    __syncthreads();
    uint32_t acc = sums[t];
    for (int i = 0; i < 256; ++i) { uint32_t x = gHist[base + i]; gHist[base + i] = acc; acc += x; }
}

// block-wide exclusive scan over 256 thread values (wave32 shfl + 8 wave sums)
__device__ __forceinline__ uint32_t block_excl_scan256(uint32_t v, uint32_t* wsums,
                                                       uint32_t& total) {
    int lane = threadIdx.x & 31, wid = threadIdx.x >> 5;
    uint32_t inc = v;
#pragma unroll
    for (int d = 1; d < 32; d <<= 1) {
        uint32_t n = __shfl_up(inc, d, 32);
        if (lane >= d) inc += n;
    }
    if (lane == 31) wsums[wid] = inc;
    __syncthreads();
    if (threadIdx.x == 0) {
        uint32_t acc = 0;
#pragma unroll
        for (int i = 0; i < 8; ++i) { uint32_t x = wsums[i]; wsums[i] = acc; acc += x; }
        wsums[8] = acc;
    }
    __syncthreads();
    total = wsums[8];
    return inc - v + wsums[wid];
}

// =====================================================================
// Stable scatter: local bit-split sort in LDS, then ranked global scatter
// =====================================================================
__global__ void radix_scatter_kernel(const uint32_t* __restrict__ lin,
                                     const uint32_t* __restrict__ valsIn,
                                     uint32_t* __restrict__ valsOut,
                                     const uint32_t* __restrict__ gScan, int shift) {
    __shared__ uint32_t sV[2][TILE];
    __shared__ uint32_t sD[2][TILE];
    __shared__ uint32_t wsums[9];
    __shared__ uint32_t digitStart[256];

    int t = threadIdx.x;
    int base = blockIdx.x * TILE;
    for (int i = t; i < TILE; i += 256) {
        uint32_t v = valsIn[base + i];
        sV[0][i] = v;
        sD[0][i] = (lin[v] >> shift) & 255u;
    }
    __syncthreads();

    int cur = 0;
    int off = t * IPT;
    for (int bit = 0; bit < 8; ++bit) {
        uint32_t bitv[IPT];
        uint32_t ones = 0;
#pragma unroll
        for (int j = 0; j < IPT; ++j) {
            bitv[j] = (sD[cur][off + j] >> bit) & 1u;
            ones += bitv[j];
        }
        uint32_t totOnes;
        uint32_t onesBefore = block_excl_scan256(ones, wsums, totOnes);
        uint32_t totalFalses = TILE - totOnes;
        int nxt = cur ^ 1;
#pragma unroll
        for (int j = 0; j < IPT; ++j) {
            int idx = off + j;
            uint32_t f = (uint32_t)idx - onesBefore;          // falses before idx
            uint32_t newPos = bitv[j] ? (totalFalses + onesBefore) : f;
            sV[nxt][newPos] = sV[cur][idx];
            sD[nxt][newPos] = sD[cur][idx];
            onesBefore += bitv[j];
        }
        __syncthreads();
        cur = nxt;
    }

    // first occurrence of each digit in the (now sorted) tile
    for (int i = t; i < TILE; i += 256) {
        uint32_t d = sD[cur][i];
        if (i == 0 || sD[cur][i - 1] != d) digitStart[d] = (uint32_t)i;
    }
    __syncthreads();
    for (int i = t; i < TILE; i += 256) {
        uint32_t d = sD[cur][i];
        uint32_t gpos = gScan[d * NBLK + blockIdx.x] + ((uint32_t)i - digitStart[d]);
        valsOut[gpos] = sV[cur][i];
    }
}

// =====================================================================
// Fused gathered-GEMM (V_WMMA_F32_16X16X4_F32) + LayerNorm + mask + new_idx
//   block = 128 threads = 4 wave32; each wave owns one 16-row strip and
//   computes all 128 output columns (8 WMMA tiles, K=64 -> 16 k-steps).
// =====================================================================
#define GWAVES 4
#define AST    68     // LDS row stride for A tile (floats): bank-free, 16B aligned
#define WST    68     // LDS row stride for W^T (floats)

__global__ __launch_bounds__(128) void spconv_wmma_ln_kernel(
    const float* __restrict__ feats,
    const int*   __restrict__ indices,
    const float* __restrict__ weight,   // [CIN][COUT] row-major
    const float* __restrict__ gamma,
    const float* __restrict__ beta,
    const uint32_t* __restrict__ order,
    float* __restrict__ out,            // [N][COUT]
    int*   __restrict__ newIdx)         // [N][4]
{
    __shared__ __attribute__((aligned(16))) float wT[COUT * WST];        // wT[n*WST + k]
    __shared__ __attribute__((aligned(16))) float aT[GWAVES][16 * AST];  // aT[w][m*AST + k]

    const int tid    = threadIdx.x;
    const int wave   = tid >> 5;
    const int lane   = tid & 31;
    const int laneLo = lane & 15;
    const int laneHi = lane >> 4;

    // stage W transposed: weight[k*COUT+n] -> wT[n*WST+k]
    for (int i = tid; i < CIN * COUT; i += 128) {
        int k = i >> 7, n = i & 127;
        wT[n * WST + k] = weight[i];
    }

    const int strip   = blockIdx.x * GWAVES + wave;
    const int rowBase = strip * 16;

    // gather this wave's 16 rows (64 floats each) into LDS; 2 lanes per row
    {
        const int r       = lane >> 1;
        const int colBase = (lane & 1) * 32;
        const uint32_t src = order[rowBase + r];
        const float4* p = (const float4*)(feats + (size_t)src * CIN + colBase);
        float4* q = (float4*)(&aT[wave][r * AST + colBase]);
#pragma unroll
        for (int j = 0; j < 8; ++j) q[j] = p[j];
    }
    __syncthreads();

    // K loop: 16 steps of V_WMMA_F32_16X16X4_F32 across 8 column tiles
    v8f acc[8] = {};
    const float* aBase = &aT[wave][laneLo * AST + 2 * laneHi];  // A: m=laneLo, k=4s+2*laneHi+{0,1}
    const float* bBase = &wT[laneLo * WST + 2 * laneHi];        // B: n=ct*16+laneLo, same k pair
    for (int s = 0; s < 16; ++s) {
        v2f a = *(const v2f*)(aBase + 4 * s);
#pragma unroll
        for (int ct = 0; ct < 8; ++ct) {
            v2f b = *(const v2f*)(bBase + ct * 16 * WST + 4 * s);
            acc[ct] = __builtin_amdgcn_wmma_f32_16x16x4_f32(
                false, a, false, b, (short)0, acc[ct], false, false);
        }
    }

    // gamma/beta per output column (n = ct*16 + laneLo)
    float gmm[8], btv[8];
#pragma unroll
    for (int ct = 0; ct < 8; ++ct) {
        gmm[ct] = gamma[ct * 16 + laneLo];
        btv[ct] = beta[ct * 16 + laneLo];
    }

    // stride mask + new indices; lane L handles row laneLo (dup in upper half)
    const uint32_t srcL = order[rowBase + laneLo];
    const int* ip = indices + 4 * (size_t)srcL;
    int ib = ip[0], itv = ip[1], iy = ip[2], ix = ip[3];
    int mok = (((iy | ix) & 1) == 0) ? 1 : 0;
    if (lane < 16) {
        int* np = newIdx + 4 * (size_t)(rowBase + laneLo);
        np[0] = mok ? ib : -1;
        np[1] = mok ? itv : -1;
        np[2] = mok ? (iy >> 1) : -1;
        np[3] = mok ? (ix >> 1) : -1;
    }

    // LayerNorm over 128 channels per row; C layout: VGPR r <-> rows r / r+8
    const float invC = 1.0f / (float)COUT;
#pragma unroll
    for (int r = 0; r < 8; ++r) {
        float s0 = 0.f, s1 = 0.f;
#pragma unroll
        for (int ct = 0; ct < 8; ++ct) { float v = acc[ct][r]; s0 += v; s1 += v * v; }
#pragma unroll
        for (int d = 1; d < 16; d <<= 1) {       // reduce within each 16-lane half
            s0 += __shfl_xor(s0, d, 32);
            s1 += __shfl_xor(s1, d, 32);
        }
        float mu  = s0 * invC;
        float var = s1 * invC - mu * mu;
        float sc  = rsqrtf(var + EPS_LN);
        float msk = (float)__shfl(mok, r + 8 * laneHi, 32);
        const int row = rowBase + r + 8 * laneHi;
        float* op = out + (size_t)row * COUT + laneLo;
#pragma unroll
        for (int ct = 0; ct < 8; ++ct) {
            float v = (acc[ct][r] - mu) * sc * gmm[ct] + btv[ct];
            op[ct * 16] = v * msk;
        }
    }
}

// =====================================================================
// Launch
// =====================================================================
extern "C" void kernel_launch(void* const* d_in, const int* in_sizes, int n_in,
                              void* d_out, int out_size, void* d_ws, size_t ws_size,
                              hipStream_t stream) {
    const float* features = (const float*)d_in[0];
    const int*   indices  = (const int*)d_in[1];
    const float* weight   = (const float*)d_in[2];
    const float* gamma    = (const float*)d_in[3];
    const float* beta     = (const float*)d_in[4];

    float* out    = (float*)d_out;
    int*   newIdx = (int*)(out + (size_t)N_VOX * COUT);

    // workspace layout: lin[N] | valA[N] | valB[N] | gHist[256*NBLK]   (~6.3 MB)
    uint32_t* lin   = (uint32_t*)d_ws;
    uint32_t* valA  = lin + N_VOX;
    uint32_t* valB  = valA + N_VOX;
    uint32_t* gHist = valB + N_VOX;

    lin_init_kernel<<<N_VOX / 256, 256, 0, stream>>>(indices, lin, valA);

    uint32_t* cur = valA;
    uint32_t* nxt = valB;
    for (int pass = 0; pass < 3; ++pass) {          // 24 bits cover the 23-bit key
        int shift = pass * 8;
        radix_hist_kernel<<<NBLK, 256, 0, stream>>>(lin, cur, gHist, shift);
        radix_scan_kernel<<<1, 256, 0, stream>>>(gHist);
        radix_scatter_kernel<<<NBLK, 256, 0, stream>>>(lin, cur, nxt, gHist, shift);
        uint32_t* tmp = cur; cur = nxt; nxt = tmp;
    }
    // after 3 passes the stable order permutation lives in `cur`

    spconv_wmma_ln_kernel<<<N_VOX / (16 * GWAVES), 128, 0, stream>>>(
        features, indices, weight, gamma, beta, cur, out, newIdx);
}